// Downsmapling_layer_with_FPS_6158983102651
// MI455X (gfx1250) — compile-verified
//
#include <hip/hip_runtime.h>

// Problem constants (from reference setup_inputs)
#define BATCH   8
#define NPTS    65536
#define CIN     128
#define COUT    256
#define NPOINT  256

typedef __attribute__((ext_vector_type(2))) float v2f;
typedef __attribute__((ext_vector_type(8))) float v8f;

// ---------------------------------------------------------------------------
// Kernel 1: Furthest Point Sampling. One 1024-thread block (32 wave32) per
// batch. Running min-distance lives in registers (64 floats/thread, fully
// unrolled). Per step: distance update + two-level argmax reduction
// (intra-wave shuffles, then wave 0 reduces 32 partials). Tie-break = lowest
// index, matching jnp.argmax.
// ---------------------------------------------------------------------------
#define FPS_THREADS 1024
#define PPT (NPTS / FPS_THREADS)   // 64 points per thread

__global__ __launch_bounds__(FPS_THREADS)
void fps_kernel(const float* __restrict__ xyz, int* __restrict__ idx_out) {
    const int b   = blockIdx.x;
    const int tid = threadIdx.x;
    const float* xb = xyz + (size_t)b * NPTS * 3;

    float dist[PPT];
#pragma unroll
    for (int j = 0; j < PPT; ++j) dist[j] = 1e10f;

    __shared__ float s_px, s_py, s_pz;
    __shared__ float s_val[32];
    __shared__ int   s_idx[32];

    if (tid == 0) {
        idx_out[b * NPOINT] = 0;
        s_px = xb[0]; s_py = xb[1]; s_pz = xb[2];
    }
    __syncthreads();

    for (int it = 1; it < NPOINT; ++it) {
        const float px = s_px, py = s_py, pz = s_pz;
        float best = -1.0f;
        int   besti = 0;
#pragma unroll
        for (int j = 0; j < PPT; ++j) {
            const int p = tid + j * FPS_THREADS;        // coalesced
            const float dx = xb[3 * p + 0] - px;
            const float dy = xb[3 * p + 1] - py;
            const float dz = xb[3 * p + 2] - pz;
            const float d  = dx * dx + dy * dy + dz * dz;
            const float dm = fminf(dist[j], d);
            dist[j] = dm;
            if (dm > best) { best = dm; besti = p; }    // strict > keeps lowest p
        }
        // intra-wave argmax (wave32)
#pragma unroll
        for (int off = 16; off > 0; off >>= 1) {
            const float ov = __shfl_down(best, off, 32);
            const int   oi = __shfl_down(besti, off, 32);
            if (ov > best || (ov == best && oi < besti)) { best = ov; besti = oi; }
        }
        const int wave = tid >> 5;
        if ((tid & 31) == 0) { s_val[wave] = best; s_idx[wave] = besti; }
        __syncthreads();
        if (wave == 0) {
            float v = s_val[tid & 31];
            int   i2 = s_idx[tid & 31];
#pragma unroll
            for (int off = 16; off > 0; off >>= 1) {
                const float ov = __shfl_down(v, off, 32);
                const int   oi = __shfl_down(i2, off, 32);
                if (ov > v || (ov == v && oi < i2)) { v = ov; i2 = oi; }
            }
            if (tid == 0) {
                idx_out[b * NPOINT + it] = i2;
                s_px = xb[3 * i2 + 0];
                s_py = xb[3 * i2 + 1];
                s_pz = xb[3 * i2 + 2];
            }
        }
        __syncthreads();
    }
}

// ---------------------------------------------------------------------------
// Kernel 2: gather features at sampled indices into contiguous feat_g
// [B, CIN, NPOINT] so the GEMM B-operand reads are linear.
// ---------------------------------------------------------------------------
__global__ void gather_kernel(const float* __restrict__ features,
                              const int* __restrict__ idx,
                              float* __restrict__ feat_g) {
    const int t = blockIdx.x * blockDim.x + threadIdx.x;   // B*CIN*NPOINT threads
    const int p = t % NPOINT;
    const int c = (t / NPOINT) % CIN;
    const int b = t / (NPOINT * CIN);
    feat_g[t] = features[((size_t)(b * CIN + c)) * NPTS + idx[b * NPOINT + p]];
}

// ---------------------------------------------------------------------------
// Kernel 3: fp32 WMMA GEMM  y[b] = W @ feat_g[b]   (bias is dropped: it is
// exactly cancelled by the following BatchNorm mean subtraction).
// One wave per 16x16 output tile; K=128 consumed 4 at a time via
// V_WMMA_F32_16X16X4_F32.
//   A (16x4 fp32): lanes 0-15 -> K=0,1 in v0,v1 ; lanes 16-31 -> K=2,3
//   B (4x16 fp32): mirrored (N = lane&15)
//   C/D (16x16 fp32, 8 VGPRs): lanes 0-15 M=r, lanes 16-31 M=8+r
// ---------------------------------------------------------------------------
__global__ __launch_bounds__(256)
void gemm_wmma_kernel(const float* __restrict__ W,
                      const float* __restrict__ feat_g,
                      float* __restrict__ y) {
    const int b     = blockIdx.y;
    const int lane  = threadIdx.x & 31;
    const int wave  = threadIdx.x >> 5;
    const int tile  = blockIdx.x * 8 + wave;        // 0..255
    const int NT_N  = NPOINT / 16;                  // 16
    const int mbase = (tile / NT_N) * 16;
    const int nbase = (tile % NT_N) * 16;

    const int row   = lane & 15;
    const int khalf = (lane >> 4) * 2;              // 0 or 2

    const float* Wrow = W + (size_t)(mbase + row) * CIN;
    const float* Fb   = feat_g + (size_t)b * CIN * NPOINT + nbase + row;

    v8f acc = {};
    for (int k = 0; k < CIN; k += 4) {
        v2f a, bv;
        a.x  = Wrow[k + khalf + 0];
        a.y  = Wrow[k + khalf + 1];
        bv.x = Fb[(size_t)(k + khalf + 0) * NPOINT];
        bv.y = Fb[(size_t)(k + khalf + 1) * NPOINT];
        acc = __builtin_amdgcn_wmma_f32_16x16x4_f32(
            /*neg_a=*/false, a, /*neg_b=*/false, bv,
            /*c_mod=*/(short)0, acc, /*reuse_a=*/false, /*reuse_b=*/false);
    }

    const int n    = nbase + (lane & 15);
    const int mofs = (lane >> 4) * 8;
#pragma unroll
    for (int r = 0; r < 8; ++r) {
        y[((size_t)b * COUT + (mbase + mofs + r)) * NPOINT + n] = acc[r];
    }
}

// ---------------------------------------------------------------------------
// Kernel 4: BatchNorm (batch stats over B*NPOINT, biased var) + affine + ReLU.
// One block per channel; 2048 values per channel.
// ---------------------------------------------------------------------------
__global__ __launch_bounds__(256)
void bn_relu_kernel(const float* __restrict__ y,
                    const float* __restrict__ gamma,
                    const float* __restrict__ beta,
                    float* __restrict__ out) {
    const int c   = blockIdx.x;
    const int tid = threadIdx.x;
    const int total = BATCH * NPOINT;               // 2048

    float s = 0.f, ss = 0.f;
    for (int i = tid; i < total; i += 256) {
        const int bb = i / NPOINT, p = i % NPOINT;
        const float v = y[((size_t)bb * COUT + c) * NPOINT + p];
        s += v; ss += v * v;
    }
    __shared__ float sv[256], svv[256];
    sv[tid] = s; svv[tid] = ss;
    __syncthreads();
    for (int off = 128; off > 0; off >>= 1) {
        if (tid < off) { sv[tid] += sv[tid + off]; svv[tid] += svv[tid + off]; }
        __syncthreads();
    }
    const float mean = sv[0] / (float)total;
    const float var  = svv[0] / (float)total - mean * mean;
    const float inv  = rsqrtf(var + 1e-5f);
    const float g = gamma[c], be = beta[c];
    for (int i = tid; i < total; i += 256) {
        const int bb = i / NPOINT, p = i % NPOINT;
        const float v = y[((size_t)bb * COUT + c) * NPOINT + p];
        const float o = (v - mean) * inv * g + be;
        out[((size_t)bb * COUT + c) * NPOINT + p] = fmaxf(o, 0.0f);
    }
}

// ---------------------------------------------------------------------------
extern "C" void kernel_launch(void* const* d_in, const int* in_sizes, int n_in,
                              void* d_out, int out_size, void* d_ws, size_t ws_size,
                              hipStream_t stream) {
    const float* xyz      = (const float*)d_in[0];
    const float* features = (const float*)d_in[1];
    const float* W        = (const float*)d_in[2];
    // d_in[3] = bias: mathematically a no-op under BatchNorm mean subtraction.
    const float* gamma    = (const float*)d_in[4];
    const float* beta     = (const float*)d_in[5];
    float* out            = (float*)d_out;

    // Workspace layout
    char* ws = (char*)d_ws;
    int*   idx    = (int*)ws;                                   // B*NPOINT ints
    float* feat_g = (float*)(ws + BATCH * NPOINT * sizeof(int));// B*CIN*NPOINT
    float* yb     = feat_g + (size_t)BATCH * CIN * NPOINT;      // B*COUT*NPOINT

    // 1) FPS: one workgroup per batch
    fps_kernel<<<BATCH, FPS_THREADS, 0, stream>>>(xyz, idx);

    // 2) Gather: B*CIN*NPOINT = 262144 elements
    gather_kernel<<<(BATCH * CIN * NPOINT) / 1024, 1024, 0, stream>>>(features, idx, feat_g);

    // 3) WMMA GEMM: (COUT/16)*(NPOINT/16)=256 tiles per batch, 8 waves/block
    gemm_wmma_kernel<<<dim3(32, BATCH), 256, 0, stream>>>(W, feat_g, yb);

    // 4) BN + ReLU: one block per output channel
    bn_relu_kernel<<<COUT, 256, 0, stream>>>(yb, gamma, beta, out);
}